// FeatureLevel_39410619908164
// MI455X (gfx1250) — compile-verified
//
#include <hip/hip_runtime.h>
#include <hip/hip_bf16.h>

// ---------------------------------------------------------------------------
// FeatureLevel gather kernel for MI455X (gfx1250, wave32).
//
// Roofline: N=2^21 points * 40 f32 out = 335 MB written, ~0.3 GFLOP total ->
// bandwidth bound (~20us floor @ 23.3 TB/s). No matrix contraction exists, so
// WMMA does not apply; optimization targets the memory system:
//   * Pre-pass packs the 4-bit fake-quant codes of 8 channels into one u32
//     per texel -> gather tables shrink from 160 MB (f32) to 21 MB, which is
//     resident in the 192 MB L2. Each gather tap = ONE global_load_b32.
//   * Streaming traffic (uv in, f32 grids in pre-pass, 335 MB out) uses
//     non-temporal TH hints so it does not evict the packed tables from L2.
//   * Dequant (code * 1/16 - 15/32) is exact in f32; bilinear path factors the
//     affine dequant out of the weighted sum: sum w*(c*q+lo) = q*S + lo*W.
//   * Branchless taps: indices clamped, validity folded into the weight ->
//     no EXEC divergence, all 8 gather loads unconditional & clause-able.
// ---------------------------------------------------------------------------

typedef float    v2f __attribute__((ext_vector_type(2)));
typedef float    v4f __attribute__((ext_vector_type(4)));
typedef unsigned v4u __attribute__((ext_vector_type(4)));

#define RES0   2048
#define RES1   1024
#define NPTS   2097152
#define QSTEP  0.0625f     //  1/16
#define QLO    (-0.46875f) // -15/32

// code = clip(rint((x - lo)/q), 0, 15) ; (x-lo)/q == 16x + 7.5
// rintf -> v_rndne_f32 (round-half-even) == jnp.round semantics.
__device__ __forceinline__ unsigned quant_code(float v) {
    float c = rintf(fmaf(v, 16.0f, 7.5f));
    c = fminf(fmaxf(c, 0.0f), 15.0f);
    return (unsigned)c;
}

// 4 nibble codes starting at bit `sh` -> floats (no dequant)
__device__ __forceinline__ v4f cvt4(unsigned q, int sh) {
    v4f r;
    r.x = (float)((q >> (sh + 0))  & 15u);
    r.y = (float)((q >> (sh + 4))  & 15u);
    r.z = (float)((q >> (sh + 8))  & 15u);
    r.w = (float)((q >> (sh + 12)) & 15u);
    return r;
}

// 4 nibble codes -> dequantized values
__device__ __forceinline__ v4f dq4(unsigned q, int sh) {
    return cvt4(q, sh) * QSTEP + QLO;
}

__device__ __forceinline__ int clampi(int v, int lo, int hi) {
    return v < lo ? lo : (v > hi ? hi : v);
}

// ---------------------------------------------------------------------------
// Pre-pass: pack 8 channel planes of f32 into one u32 of 4-bit codes/texel.
// Each thread handles 4 consecutive texels (b128 NT loads per channel plane,
// one b128 store of packed codes). f32 planes are streamed NT so they do not
// pollute L2; the packed output stays cache-resident for the gather pass.
// ---------------------------------------------------------------------------
__global__ void pack_grid_kernel(const float* __restrict__ src,
                                 unsigned* __restrict__ dst,
                                 int planeElems) {
    int t   = blockIdx.x * blockDim.x + threadIdx.x;
    int idx = t * 4;
    if (idx >= planeElems) return;
    unsigned p0 = 0, p1 = 0, p2 = 0, p3 = 0;
#pragma unroll
    for (int c = 0; c < 8; ++c) {
        const v4f v = __builtin_nontemporal_load(
            (const v4f*)(src + (size_t)c * (size_t)planeElems + idx));
        const int sh = 4 * c;
        p0 |= quant_code(v.x) << sh;
        p1 |= quant_code(v.y) << sh;
        p2 |= quant_code(v.z) << sh;
        p3 |= quant_code(v.w) << sh;
    }
    v4u pk = {p0, p1, p2, p3};
    *(v4u*)(dst + idx) = pk;   // regular (temporal) store: keep in L2
}

// ---------------------------------------------------------------------------
// Main pass: one lane per point. 4 b32 taps from packed G0 + 4 b32 taps from
// packed G1 (both L2-resident, unconditional), 10x b128 NT stores of the row.
// ---------------------------------------------------------------------------
__global__ void gather_kernel(const float* __restrict__ uv,
                              const unsigned* __restrict__ pk0,
                              const unsigned* __restrict__ pk1,
                              float* __restrict__ out) {
    const int p = blockIdx.x * blockDim.x + threadIdx.x;   // NPTS % 256 == 0
    const v2f t = __builtin_nontemporal_load((const v2f*)uv + p);
    const float u = t.x, v = t.y;

    // ---- feat0: 4-corner gather on RES0 grid (clamped indices) ----
    const float px = fmaf(u, (float)RES0, -0.5f);
    const float py = fmaf(v, (float)RES0, -0.5f);
    const int x0 = (int)fminf(fmaxf(floorf(px), 0.0f), (float)(RES0 - 2));
    const int y0 = (int)fminf(fmaxf(floorf(py), 0.0f), (float)(RES0 - 2));
    const int x1 = x0 + 1, y1 = y0 + 1;
    const unsigned q00 = pk0[(size_t)y0 * RES0 + x0];
    const unsigned q01 = pk0[(size_t)y0 * RES0 + x1];
    const unsigned q10 = pk0[(size_t)y1 * RES0 + x0];
    const unsigned q11 = pk0[(size_t)y1 * RES0 + x1];

    // ---- feat1: bilinear grid_sample on RES1 grid (zeros padding) ----
    const float ix  = fmaf(u, (float)RES1, -0.5f);
    const float iy  = fmaf(v, (float)RES1, -0.5f);
    const float fx0 = floorf(ix), fy0 = floorf(iy);
    const float wx1 = ix - fx0,   wy1 = iy - fy0;
    const float wx0 = 1.0f - wx1, wy0 = 1.0f - wy1;
    const int gx0 = (int)fx0, gy0 = (int)fy0;
    const int gx1 = gx0 + 1,  gy1 = gy0 + 1;
    // validity as 0/1 weights (zeros-padding), indices clamped for the load
    const float vx0 = (gx0 >= 0) ? 1.0f : 0.0f;           // gx0 <= 1023 always
    const float vx1 = (gx1 <= RES1 - 1) ? 1.0f : 0.0f;    // gx1 >= 0 always
    const float vy0 = (gy0 >= 0) ? 1.0f : 0.0f;
    const float vy1 = (gy1 <= RES1 - 1) ? 1.0f : 0.0f;
    const int cx0 = clampi(gx0, 0, RES1 - 1), cx1 = clampi(gx1, 0, RES1 - 1);
    const int cy0 = clampi(gy0, 0, RES1 - 1), cy1 = clampi(gy1, 0, RES1 - 1);

    const float w00 = wx0 * wy0 * (vx0 * vy0);
    const float w01 = wx1 * wy0 * (vx1 * vy0);
    const float w10 = wx0 * wy1 * (vx0 * vy1);
    const float w11 = wx1 * wy1 * (vx1 * vy1);
    const float wsum = (w00 + w01) + (w10 + w11);

    const unsigned s00 = pk1[(size_t)cy0 * RES1 + cx0];
    const unsigned s01 = pk1[(size_t)cy0 * RES1 + cx1];
    const unsigned s10 = pk1[(size_t)cy1 * RES1 + cx0];
    const unsigned s11 = pk1[(size_t)cy1 * RES1 + cx1];

    v4f accLo = cvt4(s00, 0)  * w00;
    v4f accHi = cvt4(s00, 16) * w00;
    accLo += cvt4(s01, 0)  * w01;  accHi += cvt4(s01, 16) * w01;
    accLo += cvt4(s10, 0)  * w10;  accHi += cvt4(s10, 16) * w10;
    accLo += cvt4(s11, 0)  * w11;  accHi += cvt4(s11, 16) * w11;
    const float tlo = QLO * wsum;  // sum w*(c*q+lo) = q*S + lo*sum(w)

    // ---- write 40 f32 = 10 x b128, non-temporal (pure streaming) ----
    v4f* o = (v4f*)(out + (size_t)p * 40);
    __builtin_nontemporal_store(dq4(q00, 0),  o + 0);
    __builtin_nontemporal_store(dq4(q00, 16), o + 1);
    __builtin_nontemporal_store(dq4(q01, 0),  o + 2);
    __builtin_nontemporal_store(dq4(q01, 16), o + 3);
    __builtin_nontemporal_store(dq4(q10, 0),  o + 4);
    __builtin_nontemporal_store(dq4(q10, 16), o + 5);
    __builtin_nontemporal_store(dq4(q11, 0),  o + 6);
    __builtin_nontemporal_store(dq4(q11, 16), o + 7);
    __builtin_nontemporal_store(accLo * QSTEP + tlo, o + 8);
    __builtin_nontemporal_store(accHi * QSTEP + tlo, o + 9);
}

// ---------------------------------------------------------------------------
// Fallback (only if d_ws is too small for the packed tables): gather straight
// from the f32 planes with inline fake-quant. Correct but 8x the tap traffic.
// ---------------------------------------------------------------------------
__device__ __forceinline__ float fq(float v) {
    return fmaf((float)quant_code(v), QSTEP, QLO);
}

__global__ void gather_direct_kernel(const float* __restrict__ uv,
                                     const float* __restrict__ g0,
                                     const float* __restrict__ g1,
                                     float* __restrict__ out) {
    const int p = blockIdx.x * blockDim.x + threadIdx.x;
    const v2f t = __builtin_nontemporal_load((const v2f*)uv + p);
    const float u = t.x, v = t.y;

    const float px = fmaf(u, (float)RES0, -0.5f);
    const float py = fmaf(v, (float)RES0, -0.5f);
    const int x0 = (int)fminf(fmaxf(floorf(px), 0.0f), (float)(RES0 - 2));
    const int y0 = (int)fminf(fmaxf(floorf(py), 0.0f), (float)(RES0 - 2));
    const int x1 = x0 + 1, y1 = y0 + 1;
    const size_t plane0 = (size_t)RES0 * RES0;

    float* row = out + (size_t)p * 40;
    const int xs[4] = {x0, x1, x0, x1};
    const int ys[4] = {y0, y0, y1, y1};
#pragma unroll
    for (int k = 0; k < 4; ++k) {
        const size_t base = (size_t)ys[k] * RES0 + xs[k];
#pragma unroll
        for (int c = 0; c < 8; ++c)
            row[k * 8 + c] = fq(g0[(size_t)c * plane0 + base]);
    }

    const float ix  = fmaf(u, (float)RES1, -0.5f);
    const float iy  = fmaf(v, (float)RES1, -0.5f);
    const float fx0 = floorf(ix), fy0 = floorf(iy);
    const float wx1 = ix - fx0,   wy1 = iy - fy0;
    const float wx0 = 1.0f - wx1, wy0 = 1.0f - wy1;
    const int gx0 = (int)fx0, gy0 = (int)fy0;
    const int gx1 = gx0 + 1,  gy1 = gy0 + 1;
    const float vx0 = (gx0 >= 0) ? 1.0f : 0.0f;
    const float vx1 = (gx1 <= RES1 - 1) ? 1.0f : 0.0f;
    const float vy0 = (gy0 >= 0) ? 1.0f : 0.0f;
    const float vy1 = (gy1 <= RES1 - 1) ? 1.0f : 0.0f;
    const int cx0 = clampi(gx0, 0, RES1 - 1), cx1 = clampi(gx1, 0, RES1 - 1);
    const int cy0 = clampi(gy0, 0, RES1 - 1), cy1 = clampi(gy1, 0, RES1 - 1);
    const size_t plane1 = (size_t)RES1 * RES1;

    const float ws[4] = {wx0 * wy0 * (vx0 * vy0), wx1 * wy0 * (vx1 * vy0),
                         wx0 * wy1 * (vx0 * vy1), wx1 * wy1 * (vx1 * vy1)};
    const int sxs[4] = {cx0, cx1, cx0, cx1};
    const int sys[4] = {cy0, cy0, cy1, cy1};

    float acc[8] = {0, 0, 0, 0, 0, 0, 0, 0};
#pragma unroll
    for (int k = 0; k < 4; ++k) {
        const size_t base = (size_t)sys[k] * RES1 + sxs[k];
#pragma unroll
        for (int c = 0; c < 8; ++c)
            acc[c] = fmaf(fq(g1[(size_t)c * plane1 + base]), ws[k], acc[c]);
    }
#pragma unroll
    for (int c = 0; c < 8; ++c) row[32 + c] = acc[c];
}

// ---------------------------------------------------------------------------
extern "C" void kernel_launch(void* const* d_in, const int* in_sizes, int n_in,
                              void* d_out, int out_size, void* d_ws, size_t ws_size,
                              hipStream_t stream) {
    const float* uv = (const float*)d_in[0];   // [N,2]
    const float* g0 = (const float*)d_in[1];   // [1,8,2048,2048]
    const float* g1 = (const float*)d_in[2];   // [1,8,1024,1024]
    float* out = (float*)d_out;                // [N,40]

    const size_t e0 = (size_t)RES0 * RES0;     // 4,194,304 texels
    const size_t e1 = (size_t)RES1 * RES1;     // 1,048,576 texels
    const size_t need = (e0 + e1) * sizeof(unsigned);  // ~20 MB

    if (ws_size >= need) {
        unsigned* pk0 = (unsigned*)d_ws;
        unsigned* pk1 = pk0 + e0;
        pack_grid_kernel<<<(int)(e0 / 4 / 256), 256, 0, stream>>>(g0, pk0, (int)e0);
        pack_grid_kernel<<<(int)(e1 / 4 / 256), 256, 0, stream>>>(g1, pk1, (int)e1);
        gather_kernel<<<NPTS / 256, 256, 0, stream>>>(uv, pk0, pk1, out);
    } else {
        gather_direct_kernel<<<NPTS / 256, 256, 0, stream>>>(uv, g0, g1, out);
    }
}